// NetMambaFeatureExtractor_87179246174629
// MI455X (gfx1250) — compile-verified
//
#include <hip/hip_runtime.h>
#include <math.h>

// NetMamba feature extractor, exact dependency-cone reduction:
// out[:,0,:] depends only on the (batch-uniform) cls token, so we run a
// single-token 4-layer Mamba pipeline in one workgroup (all activations in
// LDS, GEMVs via V_WMMA_F32_16X16X4_F32 tiles), then broadcast the 256-float
// result to all 256 batch rows.

#define BATCH 256
#define SEQL  50
#define DM    256   // model dim
#define NLAYER 4
#define NSTATE 16   // N
#define KCONV  4    // K
#define RRANK  16   // R
#define XPN    48   // R + 2N
#define EPSI   1e-5f

typedef float v2f __attribute__((ext_vector_type(2)));
typedef float v8f __attribute__((ext_vector_type(8)));

__device__ __forceinline__ float block_sum256(float v, float* s_red) {
#pragma unroll
  for (int off = 16; off > 0; off >>= 1) v += __shfl_xor(v, off, 32);
  const int lane = threadIdx.x & 31;
  const int wave = threadIdx.x >> 5;
  if (lane == 0) s_red[wave] = v;
  __syncthreads();
  if (wave == 0) {
    float t = (lane < 8) ? s_red[lane] : 0.0f;
#pragma unroll
    for (int off = 4; off > 0; off >>= 1) t += __shfl_xor(t, off, 32);
    if (lane == 0) s_red[0] = t;
  }
  __syncthreads();
  float total = s_red[0];
  __syncthreads();
  return total;
}

// out[n] = sum_{k<K} vec[k] * W[k*N + n], W row-major (K x N).
// K % 4 == 0, N % 16 == 0. Executed by all 8 waves of a 256-thread block;
// each wave owns 16-wide column tiles (wave-uniform loop => EXEC all-ones
// for every WMMA). A-fragment carries vec in row M=0 only; result row 0
// lives in D VGPR0 on lanes 0..15.
__device__ __forceinline__ void gemv_wmma(const float* __restrict__ W,
                                          const float* __restrict__ vec,
                                          int K, int N,
                                          float* __restrict__ out) {
  const int lane = threadIdx.x & 31;
  const int wave = threadIdx.x >> 5;
  const int m    = lane & 15;   // row (A) / column-in-tile (B, D)
  const int kg   = lane >> 4;   // K half-group: lanes 0-15 -> K{0,1}, 16-31 -> K{2,3}
  const int ntiles = N >> 4;
  for (int t = wave; t < ntiles; t += 8) {
    const int n0 = t << 4;
    v8f acc = {};
    for (int k0 = 0; k0 < K; k0 += 4) {
      const int k = k0 + 2 * kg;
      v2f a, b;
      const float a0 = vec[k];
      const float a1 = vec[k + 1];
      a.x = (m == 0) ? a0 : 0.0f;   // only row 0 of A is live (GEMV)
      a.y = (m == 0) ? a1 : 0.0f;
      b.x = W[(size_t)k * N + n0 + m];
      b.y = W[(size_t)(k + 1) * N + n0 + m];
      acc = __builtin_amdgcn_wmma_f32_16x16x4_f32(
          /*neg_a=*/false, a, /*neg_b=*/false, b,
          /*c_mod=*/(short)0, acc, /*reuse_a=*/false, /*reuse_b=*/false);
    }
    if (lane < 16) out[n0 + lane] = acc[0];  // D: M=0 -> VGPR0, lanes 0..15
  }
}

__device__ __forceinline__ float silu_f(float x) {
  return x / (1.0f + expf(-x));
}

__global__ void __launch_bounds__(256)
mamba_cls_kernel(const float* __restrict__ cls_token,
                 const float* __restrict__ pos_embed,
                 const float* __restrict__ norm_w,
                 const float* __restrict__ in_proj_w,
                 const float* __restrict__ conv_w,
                 const float* __restrict__ conv_b,
                 const float* __restrict__ x_proj_w,
                 const float* __restrict__ dt_proj_w,
                 const float* __restrict__ dt_proj_b,
                 const float* __restrict__ D_skip,
                 const float* __restrict__ out_proj_w,
                 const float* __restrict__ norm_f_w,
                 float* __restrict__ result) {
  __shared__ float s_res[DM];
  __shared__ float s_h[DM];
  __shared__ float s_hn[DM];
  __shared__ float s_xz[2 * DM];
  __shared__ float s_xc[DM];
  __shared__ float s_dbl[XPN];
  __shared__ float s_dt[DM];
  __shared__ float s_y[DM];
  __shared__ float s_red[8];

  const int d = threadIdx.x;  // one channel per thread

  // h[:,0,:] = cls_token + pos_embed[:, -1, :]  (batch-uniform)
  s_h[d]   = cls_token[d] + pos_embed[SEQL * DM + d];
  s_res[d] = 0.0f;
  __syncthreads();

  for (int i = 0; i < NLAYER; ++i) {
    // residual accumulation + RMSNorm
    float r = s_res[d] + s_h[d];
    s_res[d] = r;
    float ss  = block_sum256(r * r, s_red);
    float inv = 1.0f / sqrtf(ss * (1.0f / DM) + EPSI);
    s_hn[d] = r * inv * norm_w[i * DM + d];
    __syncthreads();

    // xz = hn @ in_proj_w  (256 -> 512)
    gemv_wmma(in_proj_w + (size_t)i * DM * 2 * DM, s_hn, DM, 2 * DM, s_xz);
    __syncthreads();

    // causal depthwise conv at t=0 (only tap K-1 hits real data) + SiLU.
    float xc = s_xz[d] * conv_w[i * DM * KCONV + d * KCONV + (KCONV - 1)]
             + conv_b[i * DM + d];
    xc = silu_f(xc);
    s_xc[d] = xc;
    float zsil = silu_f(s_xz[DM + d]);  // gate, used after the scan
    __syncthreads();

    // x_dbl = x_c @ x_proj_w  (256 -> 48 = [dt_r | B | C])
    gemv_wmma(x_proj_w + (size_t)i * DM * XPN, s_xc, DM, XPN, s_dbl);
    __syncthreads();

    // dt = softplus(x_dbl[:R] @ dt_proj_w + dt_proj_b)  (16 -> 256)
    gemv_wmma(dt_proj_w + (size_t)i * RRANK * DM, s_dbl, RRANK, DM, s_dt);
    __syncthreads();

    float dtr = s_dt[d] + dt_proj_b[i * DM + d];
    float dt  = (dtr > 20.0f) ? dtr : log1pf(expf(dtr));

    // Selective scan at t=0 with h0=0: state = dBx, so
    // y[d] = dt*x_c[d] * dot(B, C); the exp(dt*A) term multiplies zero state.
    float bc = 0.0f;
#pragma unroll
    for (int n = 0; n < NSTATE; ++n)
      bc += s_dbl[RRANK + n] * s_dbl[RRANK + NSTATE + n];

    float y = (dt * s_xc[d] * bc + D_skip[i * DM + d] * s_xc[d]) * zsil;
    s_y[d] = y;
    __syncthreads();

    // Warm next layer's in_proj weights (global_prefetch_b8).
    if (i + 1 < NLAYER)
      __builtin_prefetch(in_proj_w + (size_t)(i + 1) * DM * 2 * DM + d * 32, 0, 1);

    // h = y @ out_proj_w  (256 -> 256)
    gemv_wmma(out_proj_w + (size_t)i * DM * DM, s_y, DM, DM, s_h);
    __syncthreads();
  }

  // final RMSNorm on h (note: h, not residual), write the shared row
  float hv  = s_h[d];
  float ss  = block_sum256(hv * hv, s_red);
  float inv = 1.0f / sqrtf(ss * (1.0f / DM) + EPSI);
  result[d] = hv * inv * norm_f_w[d];
}

__global__ void __launch_bounds__(256)
broadcast_rows(const float* __restrict__ row, float* __restrict__ out) {
  out[(size_t)blockIdx.x * DM + threadIdx.x] = row[threadIdx.x];
}

extern "C" void kernel_launch(void* const* d_in, const int* in_sizes, int n_in,
                              void* d_out, int out_size, void* d_ws, size_t ws_size,
                              hipStream_t stream) {
  (void)in_sizes; (void)n_in; (void)out_size; (void)ws_size;
  // setup_inputs() order:
  // 0:x 1:proj_w 2:proj_b 3:cls_token 4:pos_embed 5:norm_w 6:in_proj_w
  // 7:conv_w 8:conv_b 9:x_proj_w 10:dt_proj_w 11:dt_proj_b 12:A_log
  // 13:D_skip 14:out_proj_w 15:norm_f_w
  // x/proj_w/proj_b/A_log are dead w.r.t. out[:,0,:] (see analysis).
  const float* cls_token  = (const float*)d_in[3];
  const float* pos_embed  = (const float*)d_in[4];
  const float* norm_w     = (const float*)d_in[5];
  const float* in_proj_w  = (const float*)d_in[6];
  const float* conv_w     = (const float*)d_in[7];
  const float* conv_b     = (const float*)d_in[8];
  const float* x_proj_w   = (const float*)d_in[9];
  const float* dt_proj_w  = (const float*)d_in[10];
  const float* dt_proj_b  = (const float*)d_in[11];
  const float* D_skip     = (const float*)d_in[13];
  const float* out_proj_w = (const float*)d_in[14];
  const float* norm_f_w   = (const float*)d_in[15];

  float* row = (float*)d_ws;  // 256 floats of scratch

  mamba_cls_kernel<<<1, 256, 0, stream>>>(
      cls_token, pos_embed, norm_w, in_proj_w, conv_w, conv_b,
      x_proj_w, dt_proj_w, dt_proj_b, D_skip, out_proj_w, norm_f_w, row);

  broadcast_rows<<<BATCH, DM, 0, stream>>>(row, (float*)d_out);
}